// Loss4_54717883351220
// MI455X (gfx1250) — compile-verified
//
#include <hip/hip_runtime.h>
#include <float.h>
#include <stdint.h>

// Problem constants (match reference)
constexpr int BROWS = 4096;   // B
constexpr int NCLS  = 50257;  // C
constexpr int TOPK  = 5;      // K
constexpr int T1    = 256;    // threads per block for phase 1 (8 wave32)

typedef __attribute__((ext_vector_type(2))) float v2f;
typedef __attribute__((ext_vector_type(4))) float v4f;
typedef __attribute__((ext_vector_type(8))) float v8f;

// ---------------------------------------------------------------------------
// Branch-filtered top-5 insert. Common case: 1 compare. Rare case: 10-op
// branchless min/max cascade that keeps t[] sorted descending.
// ---------------------------------------------------------------------------
__device__ __forceinline__ void push5(float (&t)[TOPK], float v) {
    if (v > t[TOPK - 1]) {
        float a = v;
#pragma unroll
        for (int k = 0; k < TOPK; ++k) {
            float hi = fmaxf(t[k], a);
            float lo = fminf(t[k], a);
            t[k] = hi;
            a = lo;
        }
    }
}

// ---------------------------------------------------------------------------
// Phase 1: per-row top-5 (with label masked to 0.0) + label score.
// One block per row; HBM-bandwidth-bound streaming reduction (~823 MB once).
// x is streamed with non-temporal loads: footprint (823MB) >> L2 (192MB) and
// each byte is touched exactly once, so NT avoids L2 double-handling.
// ---------------------------------------------------------------------------
__global__ __launch_bounds__(T1) void row_top5(const float* __restrict__ x,
                                               const int* __restrict__ y,
                                               float* __restrict__ m,
                                               float* __restrict__ sy) {
    const int row = blockIdx.x;
    const int tid = threadIdx.x;
    const float* __restrict__ xr = x + (size_t)row * NCLS;
    const int yi = y[row];

    float t[TOPK];
#pragma unroll
    for (int k = 0; k < TOPK; ++k) t[k] = -FLT_MAX;

    // Rows are only 4B aligned (NCLS*4 % 16 == 4): scalar lead-in to 16B.
    const uintptr_t p = (uintptr_t)xr;
    const int lead = (int)(((16u - (p & 15u)) & 15u) >> 2);

    for (int c = tid; c < lead; c += T1) {
        float v = (c == yi) ? 0.0f : xr[c];
        push5(t, v);
    }

    const v4f* __restrict__ x4 = (const v4f*)(xr + lead);
    const int nb = (NCLS - lead) >> 2;  // full float4 count
    for (int i = tid; i < nb; i += T1) {
        v4f v = __builtin_nontemporal_load(x4 + i);  // global_load_b128 th:NT
        const int base = lead + (i << 2);
        float v0 = (base + 0 == yi) ? 0.0f : v.x;
        float v1 = (base + 1 == yi) ? 0.0f : v.y;
        float v2 = (base + 2 == yi) ? 0.0f : v.z;
        float v3 = (base + 3 == yi) ? 0.0f : v.w;
        push5(t, v0);
        push5(t, v1);
        push5(t, v2);
        push5(t, v3);
    }

    for (int c = lead + (nb << 2) + tid; c < NCLS; c += T1) {
        float v = (c == yi) ? 0.0f : xr[c];
        push5(t, v);
    }

    // LDS tree merge of 256 sorted 5-lists -> block top-5.
    __shared__ float lds[T1 * TOPK];
#pragma unroll
    for (int k = 0; k < TOPK; ++k) lds[tid * TOPK + k] = t[k];
    __syncthreads();

    for (int s = T1 >> 1; s >= 1; s >>= 1) {
        if (tid < s) {
#pragma unroll
            for (int k = 0; k < TOPK; ++k) push5(t, lds[(tid + s) * TOPK + k]);
#pragma unroll
            for (int k = 0; k < TOPK; ++k) lds[tid * TOPK + k] = t[k];
        }
        __syncthreads();
    }

    if (tid == 0) {
        float sum = 0.0f;
#pragma unroll
        for (int k = 0; k < TOPK; ++k) sum += t[k];
        m[row]  = sum * (1.0f / TOPK);
        sy[row] = xr[yi];  // s_y comes from the ORIGINAL (unmasked) x
    }
}

// ---------------------------------------------------------------------------
// d_out is poisoned by the harness; zero it every launch (deterministic).
// ---------------------------------------------------------------------------
__global__ void zero_out(float* out) {
    if (blockIdx.x == 0 && threadIdx.x == 0) out[0] = 0.0f;
}

// ---------------------------------------------------------------------------
// Phase 2: mean over B x B of relu(1 + m[j] - sy[i]) via rank-2 WMMA.
//   A (16x4):  col0 = -sy[i0+r], col1 = 1, cols 2..3 = 0
//   B (4x16):  row0 = 1,         row1 = 1 + m[j0+c], rows 2..3 = 0
//   D = A x B  ->  D[r][c] = 1 + m[j0+c] - sy[i0+r]
// 32-bit 16x4 A layout: lanes 0-15 hold K=0,1 in v[0],v[1]; lanes 16-31 hold
// K=2,3 (all zero here). One wave per 16-row strip, looping over 256 column
// tiles => 256 v_wmma_f32_16x16x4_f32 per wave.
//
// m[] (16 KB) is staged into LDS ONCE per block via the CDNA5 async-copy
// path (global_load_async_to_lds_b128 + s_wait_asynccnt), so the inner loop
// has no global loads / loadcnt waits — just ds_load + cndmask + wmma.
// ---------------------------------------------------------------------------
__global__ __launch_bounds__(256) void pairwise_wmma(const float* __restrict__ m,
                                                     const float* __restrict__ sy,
                                                     float* __restrict__ out) {
    __shared__ float sm[BROWS];  // 16 KB (LDS is 320 KB/WGP)

    // Async-stage m[0..4095] -> LDS: 256 lanes x 4 iters x 16B = 16 KB.
    {
        const int tid = (int)threadIdx.x;
#pragma unroll
        for (int i = 0; i < 4; ++i) {
            const int elt = (tid + i * 256) * 4;              // float4 granule
            const unsigned loff = (unsigned)(size_t)&sm[elt]; // low 32b = LDS offset
            const float* gaddr = m + elt;
            asm volatile("global_load_async_to_lds_b128 %0, %1, off"
                         :: "v"(loff), "v"(gaddr)
                         : "memory");
        }
        asm volatile("s_wait_asynccnt 0x0" ::: "memory");
        __syncthreads();
    }

    const int wave = (int)((blockIdx.x * blockDim.x + threadIdx.x) >> 5);
    const int lane = (int)(threadIdx.x & 31);
    const int i0   = wave * 16;           // row strip owned by this wave
    const int mrow = lane & 15;
    const bool lo  = lane < 16;           // lanes 0-15 carry K=0,1; rest K=2,3 (=0)

    // A operand: loaded once per wave (16 sy values).
    const float syv = sy[i0 + mrow];      // all lanes load (hi lanes read same addrs)
    v2f a;
    a.x = lo ? -syv : 0.0f;               // A[r,0] = -sy
    a.y = lo ? 1.0f : 0.0f;               // A[r,1] = 1
    const float bx = lo ? 1.0f : 0.0f;    // B[0,c] = 1 (loop-invariant)

    float acc = 0.0f;
    for (int j0 = 0; j0 < BROWS; j0 += 16) {
        const float mv = sm[j0 + mrow];   // ds_load, no divergence, short latency
        v2f b;
        b.x = bx;
        b.y = lo ? (1.0f + mv) : 0.0f;    // B[1,c] = 1 + m
        v8f c = {};
        v8f d = __builtin_amdgcn_wmma_f32_16x16x4_f32(
            /*neg_a=*/false, a, /*neg_b=*/false, b,
            /*c_mod=*/(short)0, c, /*reuse_a=*/false, /*reuse_b=*/false);
#pragma unroll
        for (int e = 0; e < 8; ++e) acc += fmaxf(d[e], 0.0f);
    }

    // wave32 reduction
#pragma unroll
    for (int off = 16; off > 0; off >>= 1) acc += __shfl_down(acc, off, 32);

    if (lane == 0) {
        const float scale = 1.0f / ((float)BROWS * (float)BROWS);
        atomicAdd(out, acc * scale);
    }
}

// ---------------------------------------------------------------------------
// Entry point
// ---------------------------------------------------------------------------
extern "C" void kernel_launch(void* const* d_in, const int* in_sizes, int n_in,
                              void* d_out, int out_size, void* d_ws, size_t ws_size,
                              hipStream_t stream) {
    (void)in_sizes; (void)n_in; (void)out_size; (void)ws_size;

    const float* x   = (const float*)d_in[0];
    const int*   y   = (const int*)d_in[1];
    float*       out = (float*)d_out;

    float* m  = (float*)d_ws;        // [BROWS]
    float* sy = m + BROWS;           // [BROWS]

    // Phase 1: streaming top-5 per row (~823 MB read, HBM-bound ~35us)
    row_top5<<<BROWS, T1, 0, stream>>>(x, y, m, sy);

    // Zero accumulator (poisoned by harness)
    zero_out<<<1, 32, 0, stream>>>(out);

    // Phase 2: 256 waves x 256 WMMA tiles = full 4096x4096 broadcast matrix
    pairwise_wmma<<<(BROWS / 16) / 8, 256, 0, stream>>>(m, sy, out);
}